// Centernet_43155831390484
// MI455X (gfx1250) — compile-verified
//
#include <hip/hip_runtime.h>
#include <stdint.h>

// ---------------------------------------------------------------------------
// CenterNet decode for MI455X (gfx1250).
//
// Roofline: input = 16*84*128*128*4 B = 88 MB, read once -> ~3.8 us @ 23.3 TB/s.
// Heatmap planes are staged global->LDS with the Tensor Data Mover and
// double-buffered against TENSORcnt so DMA overlaps the sigmoid/NMS VALU work.
// Top-k collapses to a single 53-bit key sort:
//   key = (score_bits << 21) | (0x1FFFFF - (class<<14 | idx))
// descending key order == score desc, class asc, idx asc (lax.top_k ties),
// at both the per-class and per-batch stages.
// ---------------------------------------------------------------------------

typedef unsigned int u32x4 __attribute__((ext_vector_type(4)));
typedef int          i32x8 __attribute__((ext_vector_type(8)));
typedef int          i32x4 __attribute__((ext_vector_type(4)));

#define BATCH   16
#define CHANS   84
#define HH      128
#define WW      128
#define NCLS    80
#define TOPK    100
#define HWSZ    (HH * WW)
#define THRESH  0.3f

#define NCHUNK  4
#define ACTROWS 32          // active rows per chunk
#define BUFROWS 34          // staged rows incl. halos (max)
#define CANDCAP 2048        // expected ~1850 NMS survivors per class plane

// Issue a TDM load of an (nrows x 128) fp32 tile, global -> LDS.
// Per-wave DMA op tracked by TENSORcnt; EXEC is ignored by TDM.
__device__ __forceinline__ void tdm_issue_tile(const float* gsrc, float* lds_dst,
                                               int nrows)
{
    uint64_t gaddr = (uint64_t)(uintptr_t)gsrc;
    uint32_t laddr = (uint32_t)(uintptr_t)lds_dst;     // low 32 bits = LDS byte offset

    u32x4 g0;
    g0[0] = 1u;                                        // count=1 (valid), no gather
    g0[1] = laddr;                                     // lds_addr
    g0[2] = (uint32_t)gaddr;                           // global_addr[31:0]
    g0[3] = (uint32_t)((gaddr >> 32) & 0x01FFFFFFu)    // global_addr[56:32]
          | (2u << 30);                                // type = 2 ("image")

    i32x8 g1;
    g1[0] = (int)(2u << 16);                           // data_size = 2 -> 4 bytes
    g1[1] = (int)((uint32_t)WW << 16);                 // tensor_dim0 = 128
    g1[2] = (int)((uint32_t)nrows << 16);              // tensor_dim1 = nrows
    g1[3] = (int)((uint32_t)WW << 16);                 // tile_dim0 = 128
    g1[4] = (int)nrows;                                // tile_dim1 = nrows, tile_dim2=0
    g1[5] = (int)WW;                                   // tensor_dim0_stride = 128
    g1[6] = 0;                                         // dim1_stride (unused, 2D tile)
    g1[7] = 0;

    i32x4 g2 = {0, 0, 0, 0};                           // dims 2..4 unused
    i32x4 g3 = {0, 0, 0, 0};

#if defined(__clang_major__) && (__clang_major__ >= 23)
    i32x8 g4 = {0, 0, 0, 0, 0, 0, 0, 0};
    __builtin_amdgcn_tensor_load_to_lds(g0, g1, g2, g3, g4, 0);
#else
    __builtin_amdgcn_tensor_load_to_lds(g0, g1, g2, g3, 0);
#endif
}

// ---------------------------------------------------------------------------
// Kernel 1: one workgroup per (batch, class) plane (1280 WGs, 256 thr = 8 waves).
// Double-buffered TDM pipeline over 4 chunks of 32 active rows, sigmoid in
// place, 3x3 equality-NMS, LDS-atomic compaction, 2048-key bitonic sort,
// per-class top-100 keys to workspace.
// ---------------------------------------------------------------------------
__launch_bounds__(256)
__global__ void centernet_perclass(const float* __restrict__ in,
                                   unsigned long long* __restrict__ wsTop)
{
    __shared__ float smbuf[2][BUFROWS * WW];           // 2 x 17,408 B
    __shared__ unsigned long long cand[CANDCAP];       // 16,384 B
    __shared__ int cnt;

    const int bc  = blockIdx.x;                        // 0..1279
    const int b   = bc / NCLS;
    const int c   = bc % NCLS;
    const int tid = threadIdx.x;

    if (tid == 0) cnt = 0;

    const float* plane = in + ((size_t)b * CHANS + c) * HWSZ;

    // chunk geometry: chunk i actives rows [32i, 32i+32); staged rows r0..r1
    auto chunk_r0    = [](int ch) { return (ch == 0) ? 0 : (ACTROWS * ch - 1); };
    auto chunk_r1    = [](int ch) { return (ch == NCHUNK - 1) ? (HH - 1) : (ACTROWS * ch + ACTROWS); };

    // prologue: prime both buffers (TENSORcnt -> 2), DMA runs behind compute
    if (tid < 32) {
        tdm_issue_tile(plane + (size_t)chunk_r0(0) * WW, &smbuf[0][0],
                       chunk_r1(0) - chunk_r0(0) + 1);
        tdm_issue_tile(plane + (size_t)chunk_r0(1) * WW, &smbuf[1][0],
                       chunk_r1(1) - chunk_r0(1) + 1);
    }

    for (int chunk = 0; chunk < NCHUNK; ++chunk) {
        const int r0    = chunk_r0(chunk);
        const int nrows = chunk_r1(chunk) - r0 + 1;    // 33 or 34
        float* buf      = &smbuf[chunk & 1][0];

        if (tid < 32) {
            // oldest outstanding TDM load is this chunk's (same-wave TDM ops
            // complete in order)
            if (chunk < NCHUNK - 1) __builtin_amdgcn_s_wait_tensorcnt(1);
            else                    __builtin_amdgcn_s_wait_tensorcnt(0);
        }
        __syncthreads();                               // buffer ready for all waves

        // sigmoid in place (each thread touches only its own elements)
        for (int i = tid; i < nrows * WW; i += 256)
            buf[i] = 1.0f / (1.0f + __expf(-buf[i]));
        __syncthreads();

        // 3x3 NMS on sigmoid values (matches reference equality semantics)
        const int gbase = ACTROWS * chunk;
        const int boff  = (chunk == 0) ? 0 : 1;        // buffer row = local + boff
        for (int p = tid; p < ACTROWS * WW; p += 256) {
            const int lr   = p >> 7;
            const int x    = p & (WW - 1);
            const int grow = gbase + lr;
            const int br   = lr + boff;
            const float v  = buf[br * WW + x];
            bool ok = true;
            #pragma unroll
            for (int dy = -1; dy <= 1; ++dy) {
                const int gy = grow + dy;
                if (gy < 0 || gy >= HH) continue;
                const int by = br + dy;
                #pragma unroll
                for (int dx = -1; dx <= 1; ++dx) {
                    if (dx == 0 && dy == 0) continue;
                    const int gx = x + dx;
                    if (gx < 0 || gx >= WW) continue;
                    ok = ok && (v >= buf[by * WW + gx]);
                }
            }
            if (ok) {
                const unsigned sb  = __float_as_uint(v);       // v in (0,1): monotone bits
                const unsigned idx = (unsigned)(grow * WW + x);
                const unsigned long long key =
                    ((unsigned long long)sb << 21) |
                    (unsigned long long)(0x1FFFFFu - (((unsigned)c << 14) | idx));
                const int slot = atomicAdd(&cnt, 1);           // ds_add_rtn_u32
                if (slot < CANDCAP) cand[slot] = key;
            }
        }
        __syncthreads();                               // all reads of buf done

        // refill the freed buffer with chunk+2 while others compute chunk+1
        if ((tid < 32) && (chunk + 2 < NCHUNK)) {
            const int nc = chunk + 2;
            tdm_issue_tile(plane + (size_t)chunk_r0(nc) * WW, buf,
                           chunk_r1(nc) - chunk_r0(nc) + 1);
        }
    }

    // pad unused candidate slots with 0 (score 0 < THRESH -> zero rows downstream)
    int n = cnt; if (n > CANDCAP) n = CANDCAP;
    for (int i = n + tid; i < CANDCAP; i += 256) cand[i] = 0ull;
    __syncthreads();

    // bitonic sort, descending, N = 2048
    for (int k = 2; k <= CANDCAP; k <<= 1) {
        for (int j = k >> 1; j > 0; j >>= 1) {
            for (int t = tid; t < (CANDCAP / 2); t += 256) {
                const int i   = ((t & ~(j - 1)) << 1) | (t & (j - 1));
                const int ixj = i + j;
                const unsigned long long a  = cand[i];
                const unsigned long long b2 = cand[ixj];
                const bool desc = ((i & k) == 0);
                if (desc ? (a < b2) : (a > b2)) { cand[i] = b2; cand[ixj] = a; }
            }
            __syncthreads();
        }
    }

    // per-class top-100 keys
    unsigned long long* dst = wsTop + (size_t)bc * TOPK;
    for (int i = tid; i < TOPK; i += 256) dst[i] = cand[i];
}

// ---------------------------------------------------------------------------
// Kernel 2: one workgroup per batch. Merge 80*100 keys, bitonic sort 8192
// descending in LDS, decode top-100 into detection rows (gathering only the
// 4 reg/wh scalars each -> negligible extra HBM traffic).
// ---------------------------------------------------------------------------
__launch_bounds__(256)
__global__ void centernet_decode(const float* __restrict__ in,
                                 const unsigned long long* __restrict__ wsTop,
                                 float* __restrict__ out)
{
    __shared__ unsigned long long keys[8192];          // 64 KB
    const int b   = blockIdx.x;
    const int tid = threadIdx.x;

    const unsigned long long* src = wsTop + (size_t)b * NCLS * TOPK;  // 8000 keys
    for (int i = tid; i < 8192; i += 256)
        keys[i] = (i < NCLS * TOPK) ? src[i] : 0ull;
    __syncthreads();

    for (int k = 2; k <= 8192; k <<= 1) {
        for (int j = k >> 1; j > 0; j >>= 1) {
            for (int t = tid; t < 4096; t += 256) {
                const int i   = ((t & ~(j - 1)) << 1) | (t & (j - 1));
                const int ixj = i + j;
                const unsigned long long a  = keys[i];
                const unsigned long long b2 = keys[ixj];
                const bool desc = ((i & k) == 0);
                if (desc ? (a < b2) : (a > b2)) { keys[i] = b2; keys[ixj] = a; }
            }
            __syncthreads();
        }
    }

    if (tid < TOPK) {
        const unsigned long long key = keys[tid];
        const float score = __uint_as_float((unsigned)(key >> 21));
        float o0 = 0.f, o1 = 0.f, o2 = 0.f, o3 = 0.f, o4 = 0.f, o5 = 0.f;
        if (score >= THRESH) {
            const unsigned low = 0x1FFFFFu - (unsigned)(key & 0x1FFFFFu);
            const unsigned cls = low >> 14;
            const unsigned idx = low & 0x3FFFu;
            float ys = (float)(idx >> 7);
            float xs = (float)(idx & (WW - 1));
            const float* pb = in + (size_t)b * CHANS * HWSZ;
            const float rx = pb[(size_t)(NCLS + 0) * HWSZ + idx];
            const float ry = pb[(size_t)(NCLS + 1) * HWSZ + idx];
            const float w2 = pb[(size_t)(NCLS + 2) * HWSZ + idx] * 0.5f;
            const float h2 = pb[(size_t)(NCLS + 3) * HWSZ + idx] * 0.5f;
            xs += rx; ys += ry;
            float x0 = (xs - w2) * 4.0f, y0 = (ys - h2) * 4.0f;
            float x1 = (xs + w2) * 4.0f, y1 = (ys + h2) * 4.0f;
            x0 = fminf(fmaxf(x0, 0.f), 512.f);
            x1 = fminf(fmaxf(x1, 0.f), 512.f);
            y0 = fminf(fmaxf(y0, 0.f), 512.f);
            y1 = fminf(fmaxf(y1, 0.f), 512.f);
            o0 = x0; o1 = y0; o2 = x1; o3 = y1; o4 = score; o5 = (float)cls;
        }
        float* orow = out + ((size_t)b * TOPK + tid) * 6;
        orow[0] = o0; orow[1] = o1; orow[2] = o2;
        orow[3] = o3; orow[4] = o4; orow[5] = o5;
    }
}

// ---------------------------------------------------------------------------
extern "C" void kernel_launch(void* const* d_in, const int* in_sizes, int n_in,
                              void* d_out, int out_size, void* d_ws, size_t ws_size,
                              hipStream_t stream)
{
    const float* in = (const float*)d_in[0];
    unsigned long long* wsTop = (unsigned long long*)d_ws;  // 16*80*100*8 B = 1.0 MB

    centernet_perclass<<<BATCH * NCLS, 256, 0, stream>>>(in, wsTop);
    centernet_decode  <<<BATCH,        256, 0, stream>>>(in, wsTop, (float*)d_out);
}